// GaussianMean_24730421690465
// MI455X (gfx1250) — compile-verified
//
#include <hip/hip_runtime.h>

typedef __attribute__((ext_vector_type(2))) float v2f;
typedef __attribute__((ext_vector_type(8))) float v8f;

#define TIME_SIZE 10
#define T_DIM 8
#define E_DIM 64
#define TE (T_DIM * E_DIM)      // 512 floats per node / per edge message
#define WPB 8                   // waves per block (256 threads)

// ---------------------------------------------------------------------------
// Phase 1: zero per-node counts.
// ---------------------------------------------------------------------------
__global__ void gm_zero_counts(int* __restrict__ counts, int N) {
    int i = blockIdx.x * blockDim.x + threadIdx.x;
    if (i < N) counts[i] = 0;
}

// ---------------------------------------------------------------------------
// Phase 2: in-degree histogram (int atomics, 160K total — cheap).
// ---------------------------------------------------------------------------
__global__ void gm_hist(const int* __restrict__ edge_to, int* __restrict__ counts, int Ed) {
    int i = blockIdx.x * blockDim.x + threadIdx.x;
    if (i < Ed) atomicAdd(&counts[edge_to[i]], 1);
}

// ---------------------------------------------------------------------------
// Phase 3: exclusive scan counts -> offsets[0..N], and reset counts to 0
// (reused as fill cursors). Single block; N=10K so this is microseconds.
// ---------------------------------------------------------------------------
__global__ __launch_bounds__(256) void gm_scan(int* __restrict__ counts,
                                               int* __restrict__ offsets, int N) {
    __shared__ int part[256];
    const int t = threadIdx.x;
    const int chunk = (N + 255) / 256;
    const int lo = t * chunk;
    const int hiN = (lo + chunk < N) ? lo + chunk : N;
    int s = 0;
    for (int i = lo; i < hiN; ++i) s += counts[i];
    part[t] = s;
    __syncthreads();
    if (t == 0) {
        int run = 0;
        for (int i = 0; i < 256; ++i) { int v = part[i]; part[i] = run; run += v; }
        offsets[N] = run;
    }
    __syncthreads();
    int run = part[t];
    for (int i = lo; i < hiN; ++i) { offsets[i] = run; run += counts[i]; counts[i] = 0; }
}

// ---------------------------------------------------------------------------
// Phase 4: scatter edge ids into CSR lists (cursor = counts, zeroed by scan).
// ---------------------------------------------------------------------------
__global__ void gm_fill(const int* __restrict__ edge_to, const int* __restrict__ offsets,
                        int* __restrict__ cursor, int* __restrict__ csr, int Ed) {
    int i = blockIdx.x * blockDim.x + threadIdx.x;
    if (i < Ed) {
        int to = edge_to[i];
        int pos = offsets[to] + atomicAdd(&cursor[to], 1);
        csr[pos] = i;
    }
}

// ---------------------------------------------------------------------------
// Phase 5 (heavy): one wave32 per node. Per 16-edge tile of the node's list:
//   diff2 tile via V_WMMA_F32_16X16X4_F32:
//     A(16x4) = [x0, x1, 1, |x|^2],  B(4x16) = [-2mu0, -2mu1, |mu|^2, 1]
//     D = A*B = sum_k (x_k - mu_k)^2  exactly.
//   w = sw^2*exp(-diff2) + 1.01 staged in a per-wave LDS tile [16 edges][64 ch]
//   (computed exactly once per (edge,ch); no cross-wave sync needed —
//    single-wave producer/consumer, ordered by s_wait_dscnt).
//   Then register accumulation: lane (q=lane>>4, p=lane&15) owns
//   t in [4q,4q+4), ch in [4p,4p+4): 4 float4 accumulators, all-b128 gathers.
//   Final: out = acc/max(deg,1) + bias   (b128 coalesced stores, NO atomics).
// ---------------------------------------------------------------------------
__global__ __launch_bounds__(256) void gm_node_kernel(
        const float* __restrict__ u_l, const float* __restrict__ edge_attr,
        const int* __restrict__ edge_from,
        const float* __restrict__ mu, const float* __restrict__ sqrt_weight,
        const float* __restrict__ bias,
        const int* __restrict__ offsets, const int* __restrict__ csr,
        float* __restrict__ out, int N) {
    __shared__ float w_sh[WPB][16 * E_DIM];   // 4KB per wave
    __shared__ int   f_sh[WPB][16];

    const int lane = threadIdx.x & 31;
    const int wib  = threadIdx.x >> 5;
    const int hi   = lane >> 4;      // q: t-half      (0/1)
    const int p    = lane & 15;      // ch group / WMMA row-col index
    const int node0  = blockIdx.x * WPB + wib;
    const int nstep  = gridDim.x * WPB;

    // WMMA B operand + sw^2 for the 4 channel-column tiles (loop-invariant).
    v2f bmat[4];
    float sw2[4];
#pragma unroll
    for (int c = 0; c < 4; ++c) {
        int ch = c * 16 + p;
        float m0 = mu[ch * 2 + 0];
        float m1 = mu[ch * 2 + 1];
        bmat[c].x = hi ? (m0 * m0 + m1 * m1) : (-2.0f * m0);  // K=2 | K=0 rows
        bmat[c].y = hi ? 1.0f                : (-2.0f * m1);  // K=3 | K=1 rows
        float s = sqrt_weight[ch];
        sw2[c] = s * s;
    }
    const float4 b4 = *(const float4*)(bias + 4 * p);

    for (int node = node0; node < N; node += nstep) {
        const int base = offsets[node];
        const int deg  = offsets[node + 1] - base;

        float4 acc[4];
#pragma unroll
        for (int tt = 0; tt < 4; ++tt) acc[tt] = make_float4(0.f, 0.f, 0.f, 0.f);

        for (int ts = 0; ts < deg; ts += 16) {
            const int rem = deg - ts;
            // clamp (branchless): EXEC stays all-1s for WMMA
            const int idx = (p < rem) ? p : rem - 1;
            const int eid = csr[base + ts + idx];
            const int frm = edge_from[eid];
            f_sh[wib][p] = frm;                 // both halves write identical data

            const float* row = edge_attr + (size_t)eid * 12 + TIME_SIZE;
            const float x0 = row[0];
            const float x1 = row[1];
            v2f a;
            a.x = hi ? 1.0f                : x0;   // K=2 | K=0 cols
            a.y = hi ? (x0 * x0 + x1 * x1) : x1;   // K=3 | K=1 cols

#pragma unroll
            for (int c = 0; c < 4; ++c) {
                v8f d = {};
                d = __builtin_amdgcn_wmma_f32_16x16x4_f32(
                        false, a, false, bmat[c], (short)0, d, false, false);
                // lane holds diff2 for edge m = v + 8*hi, channel ch = 16c + p
#pragma unroll
                for (int v = 0; v < 8; ++v)
                    w_sh[wib][(v + 8 * hi) * E_DIM + c * 16 + p] =
                        sw2[c] * __expf(-d[v]) + 1.01f;
            }
            // single-wave LDS producer->consumer: wait for DS stores to land
            asm volatile("s_wait_dscnt 0x0" ::: "memory");

            const int nv = (rem < 16) ? rem : 16;
            for (int e = 0; e < nv; ++e) {
                const int frme = f_sh[wib][e];
                if (e + 1 < nv)     // hide gather latency (global_prefetch_b8)
                    __builtin_prefetch(u_l + (size_t)f_sh[wib][e + 1] * TE
                                           + (4 * hi) * E_DIM + 4 * p, 0, 1);
                const float4 wv = *(const float4*)&w_sh[wib][e * E_DIM + 4 * p];
                const float* up = u_l + (size_t)frme * TE + (4 * hi) * E_DIM + 4 * p;
#pragma unroll
                for (int tt = 0; tt < 4; ++tt) {
                    const float4 u4 = *(const float4*)(up + tt * E_DIM);
                    acc[tt].x += u4.x * wv.x;
                    acc[tt].y += u4.y * wv.y;
                    acc[tt].z += u4.z * wv.z;
                    acc[tt].w += u4.w * wv.w;
                }
            }
        }

        const float inv = 1.0f / (float)((deg > 1) ? deg : 1);
        float* op = out + (size_t)node * TE + (4 * hi) * E_DIM + 4 * p;
#pragma unroll
        for (int tt = 0; tt < 4; ++tt) {
            float4 r;
            r.x = acc[tt].x * inv + b4.x;
            r.y = acc[tt].y * inv + b4.y;
            r.z = acc[tt].z * inv + b4.z;
            r.w = acc[tt].w * inv + b4.w;
            *(float4*)(op + tt * E_DIM) = r;   // coalesced b128 store
        }
    }
}

// ---------------------------------------------------------------------------
extern "C" void kernel_launch(void* const* d_in, const int* in_sizes, int n_in,
                              void* d_out, int out_size, void* d_ws, size_t ws_size,
                              hipStream_t stream) {
    const float* u_l         = (const float*)d_in[0];
    const float* edge_attr   = (const float*)d_in[1];
    const int*   edge_from   = (const int*)d_in[2];
    const int*   edge_to     = (const int*)d_in[3];
    const float* mu          = (const float*)d_in[4];
    const float* sqrt_weight = (const float*)d_in[5];
    const float* bias        = (const float*)d_in[6];

    const int Ed = in_sizes[2];          // edges
    const int N  = in_sizes[0] / TE;     // nodes (B == 1)

    float* out = (float*)d_out;
    // workspace layout: offsets[N+1] | counts/cursor[N] | csr[Ed]  (~720 KB)
    int* offsets = (int*)d_ws;
    int* counts  = offsets + (N + 1);
    int* csr     = counts + N;

    gm_zero_counts<<<(N + 255) / 256, 256, 0, stream>>>(counts, N);
    gm_hist<<<(Ed + 255) / 256, 256, 0, stream>>>(edge_to, counts, Ed);
    gm_scan<<<1, 256, 0, stream>>>(counts, offsets, N);
    gm_fill<<<(Ed + 255) / 256, 256, 0, stream>>>(edge_to, offsets, counts, csr, Ed);

    // heavy phase: one wave32 per node, 8 waves per block
    gm_node_kernel<<<(N + WPB - 1) / WPB, 256, 0, stream>>>(
        u_l, edge_attr, edge_from, mu, sqrt_weight, bias, offsets, csr, out, N);
}